// MoE_47158740910695
// MI455X (gfx1250) — compile-verified
//
#include <hip/hip_runtime.h>

// ---------------- problem constants ----------------
constexpr int HID  = 1024;       // hidden size
constexpr int NEXP = 8;          // num experts
constexpr int IMO  = 1408;       // moe intermediate
constexpr int SIS  = 2816;       // shared intermediate
constexpr int NTOK = 2048;       // B*T tokens
constexpr int NA   = NTOK * 2;   // total (token, slot) assignments

// ---------------- tiling constants -----------------
constexpr int KB   = 32;         // K chunk staged in LDS
constexpr int ASTR = 36;         // A tile LDS row stride (floats, padded)
constexpr int BSTR = 68;         // B tile LDS row stride (floats, padded)

typedef __attribute__((ext_vector_type(2))) float v2f;
typedef __attribute__((ext_vector_type(8))) float v8f;

// D = A(16x4 f32) * B(4x16 f32) + C  -- full fp32 WMMA on gfx1250
__device__ __forceinline__ v8f wmma4(v2f a, v2f b, v8f c) {
  return __builtin_amdgcn_wmma_f32_16x16x4_f32(
      /*neg_a=*/false, a, /*neg_b=*/false, b,
      /*c_mod=*/(short)0, c, /*reuse_a=*/false, /*reuse_b=*/false);
}

// Stage a 32(row) x 32(k) A chunk into LDS from 32 gathered row pointers.
// 256 threads: each loads one float4.
__device__ __forceinline__ void load_a_tile(float* aL, const float* const* rowp,
                                            int k0, int tid) {
  int r  = tid >> 3;
  int kk = (tid & 7) * 4;
  float4 v = *(const float4*)(rowp[r] + k0 + kk);
  *(float4*)(&aL[r * ASTR + kk]) = v;   // ASTR=36 -> 16B aligned (36%4==0)
}

// Stage a 64(col) x 32(k) B chunk, transposed to bL[k][col].
// W is row-major [out_dim][ldk]; 256 threads: each loads 2 float4.
__device__ __forceinline__ void load_b_tile(float* bL, const float* W,
                                            size_t ldk, int k0, int tid) {
  int c  = tid & 63;
  int ks = (tid >> 6) * 8;
  const float* src = W + (size_t)c * ldk + k0 + ks;
  float4 v0 = *(const float4*)(src);
  float4 v1 = *(const float4*)(src + 4);
  bL[(ks + 0) * BSTR + c] = v0.x;
  bL[(ks + 1) * BSTR + c] = v0.y;
  bL[(ks + 2) * BSTR + c] = v0.z;
  bL[(ks + 3) * BSTR + c] = v0.w;
  bL[(ks + 4) * BSTR + c] = v1.x;
  bL[(ks + 5) * BSTR + c] = v1.y;
  bL[(ks + 6) * BSTR + c] = v1.z;
  bL[(ks + 7) * BSTR + c] = v1.w;
}

// One wave accumulates a 16x16 C tile over a KB-deep chunk.
// ISA fragment layouts:
//   A 16x4 f32 : lanes0-15 row=lane, K={k,k+1}; lanes16-31 row=lane-16, K={k+2,k+3}
//   B  4x16 f32: v0 lanes0-15 K=k, lanes16-31 K=k+1; v1 rows k+2,k+3
//   C/D        : vgpr j -> M = j + 8*(lane>=16), N = lane&15
__device__ __forceinline__ void mma_chunk(v8f& acc, const float* aL, const float* bL,
                                          int mo, int no, int lane) {
  int half = lane >> 4;
  int r    = lane & 15;
  const float* ap = &aL[(mo + r) * ASTR + 2 * half];
  const float* bp = &bL[half * BSTR + no + r];
#pragma unroll
  for (int k = 0; k < KB; k += 4) {
    v2f a; a.x = ap[k];            a.y = ap[k + 1];
    v2f b; b.x = bp[k * BSTR];     b.y = bp[(k + 2) * BSTR];
    acc = wmma4(a, b, acc);
  }
}

// ---------------- kernel: zero counters ----------------
__global__ void k_zero(int* __restrict__ cnt) {
  if (threadIdx.x < NEXP) cnt[threadIdx.x] = 0;
}

// ---------------- kernel: router + compaction ----------------
// 256 threads = 8 waves; one wave per token.
__global__ __launch_bounds__(256) void k_router(
    const float* __restrict__ x, const float* __restrict__ gw,
    const float* __restrict__ sgw, float* __restrict__ w2f,
    float* __restrict__ swv, int* __restrict__ cnt, int* __restrict__ list) {
  int wid  = threadIdx.x >> 5;
  int lane = threadIdx.x & 31;
  int n    = blockIdx.x * 8 + wid;
  const float* xr = x + (size_t)n * HID;

  float acc[NEXP];
#pragma unroll
  for (int e = 0; e < NEXP; ++e) acc[e] = 0.f;
  float as = 0.f;
  for (int k = lane; k < HID; k += 32) {
    float xv = xr[k];
#pragma unroll
    for (int e = 0; e < NEXP; ++e) acc[e] += xv * gw[e * HID + k];
    as += xv * sgw[k];
  }
#pragma unroll
  for (int off = 16; off > 0; off >>= 1) {
#pragma unroll
    for (int e = 0; e < NEXP; ++e) acc[e] += __shfl_down(acc[e], off);
    as += __shfl_down(as, off);
  }

  if (lane == 0) {
    float m = acc[0];
#pragma unroll
    for (int e = 1; e < NEXP; ++e) m = fmaxf(m, acc[e]);
    float s = 0.f;
    float sc[NEXP];
#pragma unroll
    for (int e = 0; e < NEXP; ++e) { sc[e] = __expf(acc[e] - m); s += sc[e]; }
    float inv = 1.f / s;
#pragma unroll
    for (int e = 0; e < NEXP; ++e) sc[e] *= inv;

    int e1 = 0; float b1 = sc[0];
#pragma unroll
    for (int e = 1; e < NEXP; ++e) if (sc[e] > b1) { b1 = sc[e]; e1 = e; }
    int e2 = (e1 == 0) ? 1 : 0; float b2 = sc[e2];
#pragma unroll
    for (int e = 0; e < NEXP; ++e)
      if (e != e1 && sc[e] > b2) { b2 = sc[e]; e2 = e; }

    w2f[n * 2 + 0] = b1;
    w2f[n * 2 + 1] = b2;
    int p1 = atomicAdd(&cnt[e1], 1);
    list[e1 * NTOK + p1] = n * 2 + 0;
    int p2 = atomicAdd(&cnt[e2], 1);
    list[e2 * NTOK + p2] = n * 2 + 1;
    swv[n] = 1.f / (1.f + __expf(-as));
  }
}

// ---------------- kernel: expert gate_up + silu*up*w ----------------
// grid (IMO/64, NTOK/32, NEXP), 256 threads = 8 waves (2x4 wave tiles of 16x16).
__global__ __launch_bounds__(256) void k_expert_gu(
    const float* __restrict__ x, const float* __restrict__ wgu,
    const float* __restrict__ w2f, const int* __restrict__ cnt,
    const int* __restrict__ list, float* __restrict__ hw) {
  __shared__ float aL[32 * ASTR];
  __shared__ float bG[KB * BSTR];
  __shared__ float bU[KB * BSTR];
  __shared__ const float* rowp[32];
  __shared__ int   n2s[32];
  __shared__ float wrow[32];

  int e = blockIdx.z, tile = blockIdx.y, jt = blockIdx.x;
  int count = cnt[e];
  int row0  = tile * 32;
  if (row0 >= count) return;
  int tid = threadIdx.x;
  if (tid < 32) {
    int idx = row0 + tid;
    int ent = list[e * NTOK + (idx < count ? idx : 0)];
    n2s[tid]  = ent;
    wrow[tid] = w2f[ent];
    rowp[tid] = x + (size_t)(ent >> 1) * HID;
  }
  __syncthreads();

  const float* WG = wgu + ((size_t)e * 2 * IMO + (size_t)jt * 64) * HID;
  const float* WU = WG + (size_t)IMO * HID;

  int wid = tid >> 5, lane = tid & 31;
  int mo = (wid >> 2) * 16, no = (wid & 3) * 16;
  v8f accG = {0.f, 0.f, 0.f, 0.f, 0.f, 0.f, 0.f, 0.f};
  v8f accU = {0.f, 0.f, 0.f, 0.f, 0.f, 0.f, 0.f, 0.f};

  for (int k0 = 0; k0 < HID; k0 += KB) {
    load_a_tile(aL, rowp, k0, tid);
    load_b_tile(bG, WG, HID, k0, tid);
    load_b_tile(bU, WU, HID, k0, tid);
    __syncthreads();
    mma_chunk(accG, aL, bG, mo, no, lane);
    mma_chunk(accU, aL, bU, mo, no, lane);
    __syncthreads();
  }

  int half = lane >> 4, r = lane & 15;
#pragma unroll
  for (int j = 0; j < 8; ++j) {
    int rl = mo + j + 8 * half;
    if (row0 + rl < count) {
      float g = accG[j], u = accU[j];
      float h = (g / (1.f + __expf(-g))) * u * wrow[rl];
      hw[(size_t)n2s[rl] * IMO + jt * 64 + no + r] = h;
    }
  }
}

// ---------------- kernel: expert down projection ----------------
// grid (HID/64, NTOK/32, NEXP); K = IMO.
__global__ __launch_bounds__(256) void k_expert_down(
    const float* __restrict__ hw, const float* __restrict__ wd,
    const int* __restrict__ cnt, const int* __restrict__ list,
    float* __restrict__ op) {
  __shared__ float aL[32 * ASTR];
  __shared__ float bL[KB * BSTR];
  __shared__ const float* rowp[32];
  __shared__ int n2s[32];

  int e = blockIdx.z, tile = blockIdx.y, ht = blockIdx.x;
  int count = cnt[e];
  int row0  = tile * 32;
  if (row0 >= count) return;
  int tid = threadIdx.x;
  if (tid < 32) {
    int idx = row0 + tid;
    int ent = list[e * NTOK + (idx < count ? idx : 0)];
    n2s[tid]  = ent;
    rowp[tid] = hw + (size_t)ent * IMO;
  }
  __syncthreads();

  const float* WD = wd + ((size_t)e * HID + (size_t)ht * 64) * IMO;

  int wid = tid >> 5, lane = tid & 31;
  int mo = (wid >> 2) * 16, no = (wid & 3) * 16;
  v8f acc = {0.f, 0.f, 0.f, 0.f, 0.f, 0.f, 0.f, 0.f};

  for (int k0 = 0; k0 < IMO; k0 += KB) {
    load_a_tile(aL, rowp, k0, tid);
    load_b_tile(bL, WD, IMO, k0, tid);
    __syncthreads();
    mma_chunk(acc, aL, bL, mo, no, lane);
    __syncthreads();
  }

  int half = lane >> 4, r = lane & 15;
#pragma unroll
  for (int j = 0; j < 8; ++j) {
    int rl = mo + j + 8 * half;
    if (row0 + rl < count)
      op[(size_t)n2s[rl] * HID + ht * 64 + no + r] = acc[j];
  }
}

// ---------------- kernel: shared expert gate_up ----------------
// grid (SIS/64, NTOK/32); K = HID.
__global__ __launch_bounds__(256) void k_shared_gu(
    const float* __restrict__ x, const float* __restrict__ wg,
    const float* __restrict__ wu, float* __restrict__ hs) {
  __shared__ float aL[32 * ASTR];
  __shared__ float bG[KB * BSTR];
  __shared__ float bU[KB * BSTR];
  __shared__ const float* rowp[32];

  int tile = blockIdx.y, jt = blockIdx.x;
  int tid = threadIdx.x;
  if (tid < 32) rowp[tid] = x + (size_t)(tile * 32 + tid) * HID;
  __syncthreads();

  const float* WG = wg + (size_t)jt * 64 * HID;
  const float* WU = wu + (size_t)jt * 64 * HID;

  int wid = tid >> 5, lane = tid & 31;
  int mo = (wid >> 2) * 16, no = (wid & 3) * 16;
  v8f accG = {0.f, 0.f, 0.f, 0.f, 0.f, 0.f, 0.f, 0.f};
  v8f accU = {0.f, 0.f, 0.f, 0.f, 0.f, 0.f, 0.f, 0.f};

  for (int k0 = 0; k0 < HID; k0 += KB) {
    load_a_tile(aL, rowp, k0, tid);
    load_b_tile(bG, WG, HID, k0, tid);
    load_b_tile(bU, WU, HID, k0, tid);
    __syncthreads();
    mma_chunk(accG, aL, bG, mo, no, lane);
    mma_chunk(accU, aL, bU, mo, no, lane);
    __syncthreads();
  }

  int half = lane >> 4, r = lane & 15;
#pragma unroll
  for (int j = 0; j < 8; ++j) {
    int rl = mo + j + 8 * half;
    int n  = tile * 32 + rl;
    float g = accG[j], u = accU[j];
    hs[(size_t)n * SIS + jt * 64 + no + r] = (g / (1.f + __expf(-g))) * u;
  }
}

// ---------------- kernel: shared expert down ----------------
// grid (HID/64, NTOK/32); K = SIS.
__global__ __launch_bounds__(256) void k_shared_down(
    const float* __restrict__ hs, const float* __restrict__ wd,
    float* __restrict__ so) {
  __shared__ float aL[32 * ASTR];
  __shared__ float bL[KB * BSTR];
  __shared__ const float* rowp[32];

  int tile = blockIdx.y, ht = blockIdx.x;
  int tid = threadIdx.x;
  if (tid < 32) rowp[tid] = hs + (size_t)(tile * 32 + tid) * SIS;
  __syncthreads();

  const float* WD = wd + (size_t)ht * 64 * SIS;

  int wid = tid >> 5, lane = tid & 31;
  int mo = (wid >> 2) * 16, no = (wid & 3) * 16;
  v8f acc = {0.f, 0.f, 0.f, 0.f, 0.f, 0.f, 0.f, 0.f};

  for (int k0 = 0; k0 < SIS; k0 += KB) {
    load_a_tile(aL, rowp, k0, tid);
    load_b_tile(bL, WD, SIS, k0, tid);
    __syncthreads();
    mma_chunk(acc, aL, bL, mo, no, lane);
    __syncthreads();
  }

  int half = lane >> 4, r = lane & 15;
#pragma unroll
  for (int j = 0; j < 8; ++j) {
    int rl = mo + j + 8 * half;
    int n  = tile * 32 + rl;
    so[(size_t)n * HID + ht * 64 + no + r] = acc[j];
  }
}

// ---------------- kernel: combine ----------------
__global__ __launch_bounds__(256) void k_combine(
    const float* __restrict__ op, const float* __restrict__ so,
    const float* __restrict__ swv, float* __restrict__ out) {
  int i = blockIdx.x * 256 + threadIdx.x;
  int n = i >> 10;           // / HID
  int h = i & (HID - 1);
  float v = op[(size_t)(n * 2) * HID + h] + op[(size_t)(n * 2 + 1) * HID + h];
  out[i] = v + swv[n] * so[i];
}

// ---------------- launch ----------------
extern "C" void kernel_launch(void* const* d_in, const int* in_sizes, int n_in,
                              void* d_out, int out_size, void* d_ws, size_t ws_size,
                              hipStream_t stream) {
  const float* x     = (const float*)d_in[0];
  const float* gw    = (const float*)d_in[1];
  const float* egu   = (const float*)d_in[2];
  const float* edown = (const float*)d_in[3];
  const float* sgate = (const float*)d_in[4];
  const float* sup   = (const float*)d_in[5];
  const float* sdown = (const float*)d_in[6];
  const float* segw  = (const float*)d_in[7];
  float* out = (float*)d_out;
  float* ws  = (float*)d_ws;

  // workspace layout (floats; all offsets multiple of 4 -> 16B aligned)
  float* W2   = ws;                               // NA = 4096
  float* SW   = ws + 4096;                        // 2048
  int*   CNT  = (int*)(ws + 6144);                // 8 (pad to 16)
  int*   LIST = (int*)(ws + 6160);                // NEXP*NTOK = 16384
  float* HW   = ws + 6160 + 16384;                // NA*IMO
  float* HS   = HW + (size_t)NA * IMO;            // NTOK*SIS
  float* OP   = HS + (size_t)NTOK * SIS;          // NA*HID
  float* SO   = OP + (size_t)NA * HID;            // NTOK*HID
  // total ~71.4 MB

  k_zero<<<1, 32, 0, stream>>>(CNT);
  k_router<<<NTOK / 8, 256, 0, stream>>>(x, gw, segw, W2, SW, CNT, LIST);

  dim3 g_gu(IMO / 64, NTOK / 32, NEXP);           // 22 x 64 x 8
  k_expert_gu<<<g_gu, 256, 0, stream>>>(x, egu, W2, CNT, LIST, HW);

  dim3 g_dn(HID / 64, NTOK / 32, NEXP);           // 16 x 64 x 8
  k_expert_down<<<g_dn, 256, 0, stream>>>(HW, edown, CNT, LIST, OP);

  dim3 g_sgu(SIS / 64, NTOK / 32);                // 44 x 64
  k_shared_gu<<<g_sgu, 256, 0, stream>>>(x, sgate, sup, HS);

  dim3 g_sdn(HID / 64, NTOK / 32);                // 16 x 64
  k_shared_down<<<g_sdn, 256, 0, stream>>>(HS, sdown, SO);

  k_combine<<<(NTOK * HID) / 256, 256, 0, stream>>>(OP, SO, SW, out);
}